// AttentionPool_1297080123643
// MI455X (gfx1250) — compile-verified
//
#include <hip/hip_runtime.h>

typedef __attribute__((ext_vector_type(2))) float v2f;
typedef __attribute__((ext_vector_type(8))) float v8f;

#define IN_C 256
#define OUT_C 128
#define NHEAD 4
#define NUM_CLASSES 1000

// workspace layout (float-element offsets); total ~1.64 MB
#define OFF_V      0          // 4*256 floats   : folded weights V[h][k]
#define OFF_CVEC   1024       // 4 floats       : folded bias c[h]
#define OFF_SEGMAX 1028       // 4000 uints     : per-(class,head) max (ordered-uint encoded)
#define OFF_SEGSUM 5028       // 4000 floats    : per-(class,head) sum of exp
#define OFF_SCORE  9028       // N*4 floats     : score, then p=exp(s-max)

// ---- order-preserving float<->uint encoding so unsigned atomicMax == float max ----
__device__ __forceinline__ unsigned enc_f32(float f) {
  unsigned u = __float_as_uint(f);
  return (u & 0x80000000u) ? ~u : (u | 0x80000000u);
}
__device__ __forceinline__ float dec_f32(unsigned e) {
  return (e & 0x80000000u) ? __uint_as_float(e ^ 0x80000000u) : __uint_as_float(~e);
}

__device__ __forceinline__ void atomic_add_f32(float* p, float v) {
  __hip_atomic_fetch_add(p, v, __ATOMIC_RELAXED, __HIP_MEMORY_SCOPE_AGENT);
}

// ---------------- kernel 0: zero output + segment accumulators ----------------
__global__ void k_init(float* __restrict__ out, int out_elems,
                       unsigned* __restrict__ segmax, float* __restrict__ segsum) {
  int i = blockIdx.x * blockDim.x + threadIdx.x;
  int stride = gridDim.x * blockDim.x;
  for (int j = i; j < out_elems; j += stride) out[j] = 0.0f;
  for (int j = i; j < NUM_CLASSES * NHEAD; j += stride) {
    segmax[j] = 0u;       // encoded: below every real encoded score
    segsum[j] = 0.0f;
  }
}

// ---------------- kernel 1: fold W_att into W_lin / b_lin ----------------
// V[h][k] = sum_o W_att[o] * W_lin[(h*128+o)][k] ; c[h] = sum_o W_att[o]*b_lin[h*128+o] + b_att
__global__ void k_fold(const float* __restrict__ W_lin, const float* __restrict__ b_lin,
                       const float* __restrict__ W_att, const float* __restrict__ b_att,
                       float* __restrict__ V, float* __restrict__ cvec) {
  int i = blockIdx.x * blockDim.x + threadIdx.x;  // 0 .. 1023
  if (i < NHEAD * IN_C) {
    int h = i / IN_C, k = i % IN_C;
    const float* wl = W_lin + (size_t)(h * OUT_C) * IN_C + k;
    float s = 0.0f;
#pragma unroll 8
    for (int o = 0; o < OUT_C; ++o) s = fmaf(W_att[o], wl[(size_t)o * IN_C], s);
    V[h * IN_C + k] = s;
  }
  if (i < NHEAD) {
    float s = b_att[0];
    for (int o = 0; o < OUT_C; ++o) s = fmaf(W_att[o], b_lin[i * OUT_C + o], s);
    cvec[i] = s;
  }
}

// ---- kernel 2: scores via V_WMMA_F32_16X16X4_F32, 64 rows (4 tiles) per wave ----
// D[16x16] = A[16xK] x B[Kx16]; B cols 0..3 = V rows (heads), cols 4..15 = 0
// (pre-zeroed in LDS so the inner loop has no masking). Row indices are clamped
// to n-1 instead of predicated so every global_load_b64 is unconditional and
// EXEC stays all-ones for the WMMAs; out-of-range rows are simply not stored.
__global__ void __launch_bounds__(256) k_score(
    const float* __restrict__ X, const int* __restrict__ y,
    const float* __restrict__ V, const float* __restrict__ cvec,
    float* __restrict__ score, unsigned* __restrict__ segmax, int n) {
  __shared__ float sB[IN_C * 16];  // [k][col] : col<4 -> V[col][k], else 0
  __shared__ float sc[NHEAD];
  for (int i = threadIdx.x; i < IN_C * 16; i += blockDim.x) {
    int k = i >> 4, col = i & 15;
    sB[i] = (col < NHEAD) ? V[col * IN_C + k] : 0.0f;
  }
  if (threadIdx.x < NHEAD) sc[threadIdx.x] = cvec[threadIdx.x];
  __syncthreads();

  const int lane = threadIdx.x & 31;
  const int wave = threadIdx.x >> 5;
  const int rowBase = (blockIdx.x * 8 + wave) * 64;  // 4 sub-tiles of 16 rows
  if (rowBase >= n) return;  // uniform per wave

  const int hsel = lane & 15;  // A row-in-subtile / B column (head)
  const int half = lane >> 4;  // which K pair this lane carries
  const int koff = half * 2;

  // clamped per-subtile row pointers (always in-bounds)
  const float* pa0 = X + (size_t)min(rowBase + 0 * 16 + hsel, n - 1) * IN_C + koff;
  const float* pa1 = X + (size_t)min(rowBase + 1 * 16 + hsel, n - 1) * IN_C + koff;
  const float* pa2 = X + (size_t)min(rowBase + 2 * 16 + hsel, n - 1) * IN_C + koff;
  const float* pa3 = X + (size_t)min(rowBase + 3 * 16 + hsel, n - 1) * IN_C + koff;
  const float* pb = sB + koff * 16 + hsel;

  v8f acc0 = {}, acc1 = {}, acc2 = {}, acc3 = {};
#pragma unroll 4
  for (int k = 0; k < IN_C; k += 4) {
    v2f b;
    b.x = pb[(k + 0) * 16];  // B[k+koff  ][hsel]
    b.y = pb[(k + 1) * 16];  // B[k+koff+1][hsel]
    v2f a0 = *(const v2f*)(pa0 + k);
    v2f a1 = *(const v2f*)(pa1 + k);
    v2f a2 = *(const v2f*)(pa2 + k);
    v2f a3 = *(const v2f*)(pa3 + k);
    acc0 = __builtin_amdgcn_wmma_f32_16x16x4_f32(false, a0, false, b, (short)0, acc0, false, false);
    acc1 = __builtin_amdgcn_wmma_f32_16x16x4_f32(false, a1, false, b, (short)0, acc1, false, false);
    acc2 = __builtin_amdgcn_wmma_f32_16x16x4_f32(false, a2, false, b, (short)0, acc2, false, false);
    acc3 = __builtin_amdgcn_wmma_f32_16x16x4_f32(false, a3, false, b, (short)0, acc3, false, false);
  }

  // D layout: vgpr j -> row j (lanes 0-15) / row j+8 (lanes 16-31), col = lane%16
  if (hsel < NHEAD) {
    const float ch = sc[hsel];
    v8f accs[4] = {acc0, acc1, acc2, acc3};
#pragma unroll
    for (int t = 0; t < 4; ++t) {
#pragma unroll
      for (int j = 0; j < 8; ++j) {
        int r = rowBase + t * 16 + j + half * 8;
        if (r < n) {
          float s = accs[t][j] + ch;
          s = (s >= 0.0f) ? s : 0.2f * s;  // leaky_relu(0.2)
          score[(size_t)r * NHEAD + hsel] = s;
          atomicMax(&segmax[y[r] * NHEAD + hsel], enc_f32(s));
        }
      }
    }
  }
}

// ---------------- kernel 3: p = exp(s - segmax), seg sum ----------------
__global__ void k_exp(const int* __restrict__ y, float* __restrict__ score,
                      const unsigned* __restrict__ segmax, float* __restrict__ segsum, int n) {
  int i = blockIdx.x * blockDim.x + threadIdx.x;
  if (i >= n * NHEAD) return;
  int r = i >> 2, h = i & 3;
  int seg = y[r] * NHEAD + h;
  float p = __expf(score[i] - dec_f32(segmax[seg]));
  score[i] = p;
  atomic_add_f32(&segsum[seg], p);
}

// ---------------- kernel 4: weighted scatter-add pooling ----------------
// One block per row: 256 lanes cover IN_C columns; 4 L2-resident f32 atomics each.
__global__ void __launch_bounds__(256) k_pool(
    const float* __restrict__ X, const int* __restrict__ y,
    const float* __restrict__ p, const float* __restrict__ segsum,
    float* __restrict__ out, int n) {
  int r = blockIdx.x;
  if (r >= n) return;
  int yv = y[r];
  float w0 = p[(size_t)r * NHEAD + 0] / segsum[yv * NHEAD + 0];
  float w1 = p[(size_t)r * NHEAD + 1] / segsum[yv * NHEAD + 1];
  float w2 = p[(size_t)r * NHEAD + 2] / segsum[yv * NHEAD + 2];
  float w3 = p[(size_t)r * NHEAD + 3] / segsum[yv * NHEAD + 3];
  int t = threadIdx.x;
  float x = X[(size_t)r * IN_C + t];
  float* o = out + (size_t)yv * (NHEAD * IN_C) + t;
  atomic_add_f32(o + 0 * IN_C, w0 * x);
  atomic_add_f32(o + 1 * IN_C, w1 * x);
  atomic_add_f32(o + 2 * IN_C, w2 * x);
  atomic_add_f32(o + 3 * IN_C, w3 * x);
}

extern "C" void kernel_launch(void* const* d_in, const int* in_sizes, int n_in,
                              void* d_out, int out_size, void* d_ws, size_t ws_size,
                              hipStream_t stream) {
  const float* context_h = (const float*)d_in[0];
  const float* W_lin     = (const float*)d_in[1];
  const float* b_lin     = (const float*)d_in[2];
  const float* W_att     = (const float*)d_in[3];
  const float* b_att     = (const float*)d_in[4];
  const int*   y         = (const int*)d_in[5];
  // d_in[6] = num_classes scalar; fixed at compile time (1000)

  const int n = in_sizes[0] / IN_C;

  float*    ws     = (float*)d_ws;
  float*    V      = ws + OFF_V;
  float*    cvec   = ws + OFF_CVEC;
  unsigned* segmax = (unsigned*)(ws + OFF_SEGMAX);
  float*    segsum = ws + OFF_SEGSUM;
  float*    score  = ws + OFF_SCORE;
  float*    out    = (float*)d_out;

  k_init<<<256, 256, 0, stream>>>(out, out_size, segmax, segsum);
  k_fold<<<4, 256, 0, stream>>>(W_lin, b_lin, W_att, b_att, V, cvec);
  int tiles64 = (n + 63) / 64;
  k_score<<<(tiles64 + 7) / 8, 256, 0, stream>>>(context_h, y, V, cvec, score, segmax, n);
  k_exp<<<(n * NHEAD + 255) / 256, 256, 0, stream>>>(y, score, segmax, segsum, n);
  k_pool<<<n, 256, 0, stream>>>(context_h, y, score, segsum, out, n);
}